// GAT_36816459661698
// MI455X (gfx1250) — compile-verified
//
#include <hip/hip_runtime.h>
#include <hip/hip_bf16.h>

#define NN   100000
#define EE   1600000
#define EP   1700000   // E + N self loops
#define FIN  128
#define HH   64
#define CC   16
#define SLOPE_ATT 0.2f
#define SLOPE_ACT 0.01f

typedef float v2f __attribute__((ext_vector_type(2)));
typedef float v8f __attribute__((ext_vector_type(8)));

// ---------------- order-preserving float<->int encoding for atomicMax -------
__device__ __forceinline__ int f2ord(float f) {
    int i = __float_as_int(f);
    return (i >= 0) ? i : (i ^ 0x7fffffff);
}
__device__ __forceinline__ float ord2f(int i) {
    int j = (i >= 0) ? i : (i ^ 0x7fffffff);
    return __int_as_float(j);
}
__device__ __forceinline__ float lrelu(float v, float s) { return v > 0.f ? v : s * v; }

// ---------------- WMMA GEMM: out[M,NCOLS] = A[M,K] @ B[K,NCOLS] (+bias,+act) --
// One wave computes a 16-row tile across all NCOLS columns using
// V_WMMA_F32_16X16X4_F32. B is staged in LDS.
template <int K, int NCOLS, bool BIAS, bool ACT>
__global__ void wmma_gemm(const float* __restrict__ A, const float* __restrict__ B,
                          const float* __restrict__ bias, float* __restrict__ out, int M)
{
    constexpr int NT = NCOLS / 16;
    __shared__ float sB[K * NCOLS];
    for (int i = threadIdx.x; i < K * NCOLS; i += blockDim.x) sB[i] = B[i];
    __syncthreads();

    const int wave = threadIdx.x >> 5;
    const int lane = threadIdx.x & 31;
    const int tile = blockIdx.x * (blockDim.x >> 5) + wave;
    if (tile * 16 >= M) return;                 // wave-uniform: EXEC stays all-1

    const int row0 = tile * 16;
    const int l15  = lane & 15;
    const bool hi  = lane >= 16;                // lanes 16-31 carry K+2,K+3
    const float* Arow = A + (size_t)(row0 + l15) * K;

    v8f acc[NT] = {};
    for (int k = 0; k < K; k += 4) {
        const int kk = k + (hi ? 2 : 0);
        v2f a; a.x = Arow[kk]; a.y = Arow[kk + 1];
#pragma unroll
        for (int t = 0; t < NT; ++t) {
            const int col = t * 16 + l15;
            v2f b; b.x = sB[kk * NCOLS + col]; b.y = sB[(kk + 1) * NCOLS + col];
            acc[t] = __builtin_amdgcn_wmma_f32_16x16x4_f32(
                false, a, false, b, (short)0, acc[t], false, false);
        }
    }
#pragma unroll
    for (int t = 0; t < NT; ++t) {
        const int col = t * 16 + l15;
        const float bv = BIAS ? bias[col] : 0.f;
#pragma unroll
        for (int r = 0; r < 8; ++r) {
            const int row = row0 + r + (hi ? 8 : 0);
            float v = acc[t][r] + bv;
            if (ACT) v = lrelu(v, SLOPE_ACT);
            out[(size_t)row * NCOLS + col] = v;
        }
    }
}

// ---------------- per-node attention projections -----------------------------
__global__ void alpha_kernel(const float* __restrict__ h, const float* __restrict__ asrc,
                             const float* __restrict__ adst,
                             float* __restrict__ os, float* __restrict__ od)
{
    const int n = blockIdx.x * blockDim.x + threadIdx.x;
    if (n >= NN) return;
    const float4* hp = (const float4*)(h + (size_t)n * HH);
    float s = 0.f, d = 0.f;
#pragma unroll
    for (int i = 0; i < HH / 4; ++i) {
        const float4 v = hp[i];
        s += v.x * asrc[4*i] + v.y * asrc[4*i+1] + v.z * asrc[4*i+2] + v.w * asrc[4*i+3];
        d += v.x * adst[4*i] + v.y * adst[4*i+1] + v.z * adst[4*i+2] + v.w * adst[4*i+3];
    }
    os[n] = s; od[n] = d;
}

// ---------------- per-layer init: zero agg, reset segment stats --------------
__global__ void init_layer(float* __restrict__ agg, int* __restrict__ mord,
                           float* __restrict__ denom)
{
    const int i = blockIdx.x * blockDim.x + threadIdx.x;
    if (i < NN * HH) agg[i] = 0.f;
    if (i < NN) { mord[i] = INT_MIN; denom[i] = 0.f; }
}

__device__ __forceinline__ void edge_pair(const int* __restrict__ ei, int e,
                                          int& src, int& dst)
{
    if (e < EE) { src = ei[e]; dst = ei[EE + e]; }
    else        { src = e - EE; dst = e - EE; }   // self loops
}

// ---------------- pass 1: segment max of attention logits --------------------
__global__ void edge_max(const int* __restrict__ ei, const float* __restrict__ as,
                         const float* __restrict__ ad, int* __restrict__ mord)
{
    const int e = blockIdx.x * blockDim.x + threadIdx.x;
    if (e >= EP) return;
    int src, dst; edge_pair(ei, e, src, dst);
    const float v = lrelu(as[src] + ad[dst], SLOPE_ATT);
    atomicMax(&mord[dst], f2ord(v));
}

// ---------------- pass 2: exp + segment sum ----------------------------------
__global__ void edge_exp(const int* __restrict__ ei, const float* __restrict__ as,
                         const float* __restrict__ ad, const int* __restrict__ mord,
                         float* __restrict__ denom, float* __restrict__ exbuf)
{
    const int e = blockIdx.x * blockDim.x + threadIdx.x;
    if (e >= EP) return;
    int src, dst; edge_pair(ei, e, src, dst);
    const float v  = lrelu(as[src] + ad[dst], SLOPE_ATT);
    const float ex = expf(v - ord2f(mord[dst]));
    exbuf[e] = ex;
    atomicAdd(&denom[dst], ex);
}

// ---------------- pass 3: weighted feature scatter (32 lanes / edge) ---------
__global__ void edge_scatter(const int* __restrict__ ei, const float* __restrict__ exbuf,
                             const float* __restrict__ denom, const float* __restrict__ hlin,
                             float* __restrict__ agg)
{
    const long long tid = (long long)blockIdx.x * blockDim.x + threadIdx.x;
    const int e = (int)(tid >> 5);
    if (e >= EP) return;
    const int lane = (int)(tid & 31);
    int src, dst; edge_pair(ei, e, src, dst);
    const float w = exbuf[e] / (denom[dst] + 1e-16f);
    const float2 hv = ((const float2*)(hlin + (size_t)src * HH))[lane];
    float* ap = agg + (size_t)dst * HH + 2 * lane;
    atomicAdd(ap,     hv.x * w);
    atomicAdd(ap + 1, hv.y * w);
}

// ---------------- bias + activation into next-layer h ------------------------
__global__ void finalize(float* __restrict__ h, const float* __restrict__ b)
{
    const int i = blockIdx.x * blockDim.x + threadIdx.x;
    if (i >= NN * HH) return;
    h[i] = lrelu(h[i] + b[i & (HH - 1)], SLOPE_ACT);
}

// ---------------- final log-softmax over 16 classes --------------------------
__global__ void log_softmax_k(const float* __restrict__ logits, float* __restrict__ out)
{
    const int n = blockIdx.x * blockDim.x + threadIdx.x;
    if (n >= NN) return;
    float l[CC];
    float m = -3.4e38f;
#pragma unroll
    for (int c = 0; c < CC; ++c) { l[c] = logits[(size_t)n * CC + c]; m = fmaxf(m, l[c]); }
    float s = 0.f;
#pragma unroll
    for (int c = 0; c < CC; ++c) s += expf(l[c] - m);
    const float ls = m + logf(s);
#pragma unroll
    for (int c = 0; c < CC; ++c) out[(size_t)n * CC + c] = l[c] - ls;
}

extern "C" void kernel_launch(void* const* d_in, const int* in_sizes, int n_in,
                              void* d_out, int out_size, void* d_ws, size_t ws_size,
                              hipStream_t stream)
{
    const float* x     = (const float*)d_in[0];
    const int*   ei    = (const int*)  d_in[1];
    // d_in[2] edge_weight: unused by the reference GAT layers
    const float* W_in  = (const float*)d_in[3];
    const float* b_in  = (const float*)d_in[4];
    const float* Wc    = (const float*)d_in[5];   // [3,64,64]
    const float* a_src = (const float*)d_in[6];   // [3,64]
    const float* a_dst = (const float*)d_in[7];
    const float* bc    = (const float*)d_in[8];
    const float* W_out = (const float*)d_in[9];
    const float* b_out = (const float*)d_in[10];
    float* out = (float*)d_out;

    // workspace carve-up
    float* hA    = (float*)d_ws;                    // N*64 : h / agg
    float* hB    = hA + (size_t)NN * HH;            // N*64 : h_lin / logits
    float* als   = hB + (size_t)NN * HH;            // N
    float* ald   = als + NN;                        // N
    float* denom = ald + NN;                        // N
    int*   mord  = (int*)(denom + NN);              // N
    float* exbuf = (float*)(mord + NN);             // EP

    const int tiles      = NN / 16;                 // 6250
    const int gemmBlocks = (tiles + 3) / 4;         // 4 waves / block
    const int nThr       = (NN + 255) / 256;
    const int nhThr      = (NN * HH + 255) / 256;
    const int eThr       = (EP + 255) / 256;
    const long long scatterThreads = (long long)EP * 32;
    const int sBlocks    = (int)((scatterThreads + 255) / 256);

    // input projection: hA = leaky_relu(x @ W_in + b_in)
    wmma_gemm<FIN, HH, true, true><<<gemmBlocks, 128, 0, stream>>>(x, W_in, b_in, hA, NN);

    for (int l = 0; l < 3; ++l) {
        // hB = hA @ Wc[l]
        wmma_gemm<HH, HH, false, false><<<gemmBlocks, 128, 0, stream>>>(
            hA, Wc + (size_t)l * HH * HH, nullptr, hB, NN);
        alpha_kernel<<<nThr, 256, 0, stream>>>(hB, a_src + l * HH, a_dst + l * HH, als, ald);
        init_layer<<<nhThr, 256, 0, stream>>>(hA, mord, denom);       // hA reused as agg
        edge_max    <<<eThr, 256, 0, stream>>>(ei, als, ald, mord);
        edge_exp    <<<eThr, 256, 0, stream>>>(ei, als, ald, mord, denom, exbuf);
        edge_scatter<<<sBlocks, 256, 0, stream>>>(ei, exbuf, denom, hB, hA);
        finalize    <<<nhThr, 256, 0, stream>>>(hA, bc + l * HH);
    }

    // logits = hA @ W_out + b_out  (into hB), then log_softmax
    wmma_gemm<HH, CC, true, false><<<gemmBlocks, 128, 0, stream>>>(hA, W_out, b_out, hB, NN);
    log_softmax_k<<<nThr, 256, 0, stream>>>(hB, out);
}